// OursLSTM_86955907875350
// MI455X (gfx1250) — compile-verified
//
#include <hip/hip_runtime.h>
#include <math.h>

typedef __attribute__((ext_vector_type(2))) float v2f;
typedef __attribute__((ext_vector_type(8))) float v8f;

#define HID 256
#define BN 10400            // 32 * 325
#define NNODE 325
#define TSTEPS 12
#define KCHUNK 64
#define SH_STRIDE 33        // odd stride -> conflict-free LDS banks

__device__ __forceinline__ float sigmoidf_(float x) {
    return 1.0f / (1.0f + __expf(-x));
}

// ---------------- prep kernels ----------------

// out[r*256+n] = sum_{q<5} in[(q*cin + r)*256 + n]   (collapse identity graph-conv)
__global__ void sum_blocks_kernel(const float* __restrict__ in, float* __restrict__ out, int cin) {
    int idx = blockIdx.x * blockDim.x + threadIdx.x;
    int n = cin * HID;
    if (idx >= n) return;
    float s = 0.f;
    #pragma unroll
    for (int q = 0; q < 5; ++q) s += in[(size_t)q * n + idx];
    out[idx] = s;
}

// in [2][1024][256] -> out [2][256][1024]
__global__ void transpose2_kernel(const float* __restrict__ in, float* __restrict__ out) {
    int idx = blockIdx.x * blockDim.x + threadIdx.x;
    if (idx >= 2 * 1024 * 256) return;
    int l = idx >> 18;            // /262144
    int rem = idx & 262143;
    int r = rem >> 8;             // 0..1023 (gate row)
    int cc = rem & 255;           // 0..255  (k)
    out[(size_t)l * 262144 + (size_t)cc * 1024 + r] = in[idx];
}

__global__ void add_bias_kernel(const float* __restrict__ a, const float* __restrict__ b,
                                float* __restrict__ out, int n) {
    int i = blockIdx.x * blockDim.x + threadIdx.x;
    if (i < n) out[i] = a[i] + b[i];
}

__global__ void zero_kernel(float* __restrict__ p, int n) {
    int i = blockIdx.x * blockDim.x + threadIdx.x;
    if (i < n) p[i] = 0.f;
}

// ---------------- tiny dense layers ----------------

// encoder layer0: inputs [B,T,N,2] -> xa [BN,256] in seq order (row = b*325+node), relu
__global__ void enc_l0_kernel(const float* __restrict__ in, int t,
                              const float* __restrict__ W, const float* __restrict__ bias,
                              float* __restrict__ out) {
    int idx = blockIdx.x * blockDim.x + threadIdx.x;   // BN*256 threads
    int m = idx >> 8;
    int nn = idx & 255;
    int b = m / NNODE;
    int node = m - b * NNODE;
    size_t base = (((size_t)b * TSTEPS + t) * NNODE + node) * 2;
    float v = in[base] * W[nn] + in[base + 1] * W[HID + nn] + bias[nn];
    out[idx] = fmaxf(v, 0.f);
}

// decoder layer0: x [BN,1] -> out [BN,256], relu
__global__ void dec_l0_kernel(const float* __restrict__ x, const float* __restrict__ W,
                              const float* __restrict__ bias, float* __restrict__ out) {
    int idx = blockIdx.x * blockDim.x + threadIdx.x;
    int m = idx >> 8;
    int nn = idx & 255;
    float v = x[m] * W[nn] + bias[nn];
    out[idx] = fmaxf(v, 0.f);
}

// ---------------- WMMA GEMM: C = relu(A[M,256] @ B[256,256] + bias) ----------------
__global__ __launch_bounds__(256) void gemm_relu_kernel(
    const float* __restrict__ A, const float* __restrict__ Bm,
    const float* __restrict__ bias, float* __restrict__ C) {
    __shared__ float sh[KCHUNK * SH_STRIDE];
    const int tid = threadIdx.x;
    const int wave = tid >> 5;
    const int lane = tid & 31;
    const int half = lane >> 4;
    const int l = lane & 15;
    const int row_tile = wave >> 2;       // 0..1
    const int col_tile = wave & 3;        // 0..3
    const int m0 = blockIdx.x * 32;
    const int n0 = blockIdx.y * 64;
    const int rowA = row_tile * 16 + l;   // LDS row 0..31
    const int colC = n0 + col_tile * 16 + l;

    // hoisted base pointers -> inner loops use compile-time instruction offsets only
    const float* ap = A + (size_t)(m0 + (tid >> 6)) * HID + (tid & 63);      // cooperative fill
    float* shw = sh + (tid & 63) * SH_STRIDE + (tid >> 6);
    const float* shr = sh + rowA + (half ? 2 * SH_STRIDE : 0);               // a-frag read base
    const float* bp = Bm + colC + (half ? 2 * HID : 0);                      // b-frag read base

    v8f acc = {};
    for (int kc = 0; kc < HID; kc += KCHUNK) {
        __syncthreads();
        #pragma unroll
        for (int q = 0; q < 8; ++q)                // 32 rows x 64 k / 256 threads
            shw[4 * q] = ap[q * 4 * HID];
        __syncthreads();
        #pragma unroll
        for (int kk = 0; kk < KCHUNK; kk += 4) {
            v2f a; a.x = shr[kk * SH_STRIDE]; a.y = shr[(kk + 1) * SH_STRIDE];
            v2f b; b.x = bp[kk * HID];        b.y = bp[kk * HID + HID];
            acc = __builtin_amdgcn_wmma_f32_16x16x4_f32(false, a, false, b, (short)0, acc,
                                                        false, false);
        }
        ap += KCHUNK;
        bp += KCHUNK * HID;
    }
    float bv = bias[colC];
    float* cp = C + (size_t)(m0 + row_tile * 16 + (half ? 8 : 0)) * HID + colC;
    #pragma unroll
    for (int r = 0; r < 8; ++r)
        cp[r * HID] = fmaxf(acc[r] + bv, 0.f);
}

// ---------------- fused WMMA LSTM layer step ----------------
// gates = x@wihT + h_in@whhT + bias ; elementwise i,f,g,o update of h_out/c (in-place c)
__global__ __launch_bounds__(256) void lstm_step_kernel(
    const float* __restrict__ x, const float* __restrict__ h_in,
    float* __restrict__ h_out, float* __restrict__ c_st,
    const float* __restrict__ wihT, const float* __restrict__ whhT,
    const float* __restrict__ bias) {
    __shared__ float sh[KCHUNK * SH_STRIDE];
    const int tid = threadIdx.x;
    const int wave = tid >> 5;
    const int lane = tid & 31;
    const int half = lane >> 4;
    const int l = lane & 15;
    const int row_tile = wave >> 2;
    const int col_tile = wave & 3;
    const int m0 = blockIdx.x * 32;
    const int n0 = blockIdx.y * 64;
    const int rowA = row_tile * 16 + l;
    const int colC = n0 + col_tile * 16 + l;     // hidden column 0..255
    const int NG = 1024;

    float* shw = sh + (tid & 63) * SH_STRIDE + (tid >> 6);
    const float* shr = sh + rowA + (half ? 2 * SH_STRIDE : 0);
    const size_t fill_off = (size_t)(m0 + (tid >> 6)) * HID + (tid & 63);
    const int bcol = colC + (half ? 2 * NG : 0);

    v8f acc[4] = {{}, {}, {}, {}};
    for (int src = 0; src < 2; ++src) {
        const float* ap = (src ? h_in : x) + fill_off;
        const float* Bsel = src ? whhT : wihT;
        const float* bg0 = Bsel + bcol;
        const float* bg1 = Bsel + bcol + HID;
        const float* bg2 = Bsel + bcol + 2 * HID;
        const float* bg3 = Bsel + bcol + 3 * HID;
        for (int kc = 0; kc < HID; kc += KCHUNK) {
            __syncthreads();
            #pragma unroll
            for (int q = 0; q < 8; ++q)
                shw[4 * q] = ap[q * 4 * HID];
            __syncthreads();
            #pragma unroll
            for (int kk = 0; kk < KCHUNK; kk += 4) {
                v2f a; a.x = shr[kk * SH_STRIDE]; a.y = shr[(kk + 1) * SH_STRIDE];
                v2f b0; b0.x = bg0[kk * NG]; b0.y = bg0[kk * NG + NG];
                acc[0] = __builtin_amdgcn_wmma_f32_16x16x4_f32(false, a, false, b0, (short)0,
                                                               acc[0], false, false);
                v2f b1; b1.x = bg1[kk * NG]; b1.y = bg1[kk * NG + NG];
                acc[1] = __builtin_amdgcn_wmma_f32_16x16x4_f32(false, a, false, b1, (short)0,
                                                               acc[1], false, false);
                v2f b2; b2.x = bg2[kk * NG]; b2.y = bg2[kk * NG + NG];
                acc[2] = __builtin_amdgcn_wmma_f32_16x16x4_f32(false, a, false, b2, (short)0,
                                                               acc[2], false, false);
                v2f b3; b3.x = bg3[kk * NG]; b3.y = bg3[kk * NG + NG];
                acc[3] = __builtin_amdgcn_wmma_f32_16x16x4_f32(false, a, false, b3, (short)0,
                                                               acc[3], false, false);
            }
            ap += KCHUNK;
            bg0 += KCHUNK * NG; bg1 += KCHUNK * NG; bg2 += KCHUNK * NG; bg3 += KCHUNK * NG;
        }
    }
    float bi = bias[colC], bf = bias[HID + colC], bg = bias[2 * HID + colC], bo = bias[3 * HID + colC];
    const size_t eoff = (size_t)(m0 + row_tile * 16 + (half ? 8 : 0)) * HID + colC;
    float* cpp = c_st + eoff;
    float* hpp = h_out + eoff;
    #pragma unroll
    for (int r = 0; r < 8; ++r) {
        float iv = sigmoidf_(acc[0][r] + bi);
        float fv = sigmoidf_(acc[1][r] + bf);
        float gv = tanhf(acc[2][r] + bg);
        float ov = sigmoidf_(acc[3][r] + bo);
        float cn = fv * cpp[r * HID] + iv * gv;
        cpp[r * HID] = cn;
        hpp[r * HID] = ov * tanhf(cn);
    }
}

// ---------------- projection: out = xb @ proj_w + proj_b ; scatter to d_out + feedback ----------------
__global__ __launch_bounds__(256) void proj_kernel(const float* __restrict__ xb,
                                                   const float* __restrict__ pw,
                                                   const float* __restrict__ pb,
                                                   float* __restrict__ out, float* __restrict__ fb,
                                                   int t) {
    int wave = threadIdx.x >> 5;
    int lane = threadIdx.x & 31;
    int m = blockIdx.x * 8 + wave;
    const float* xp = xb + (size_t)m * HID + lane;
    float s = 0.f;
    #pragma unroll
    for (int q = 0; q < 8; ++q)
        s += xp[32 * q] * pw[lane + 32 * q];
    #pragma unroll
    for (int off = 16; off > 0; off >>= 1) s += __shfl_xor(s, off, 32);
    if (lane == 0) {
        float v = s + pb[0];
        int b = m / NNODE;
        int node = m - b * NNODE;
        out[((size_t)b * TSTEPS + t) * NNODE + node] = v;
        fb[m] = v;
    }
}

// ---------------- host orchestration ----------------
extern "C" void kernel_launch(void* const* d_in, const int* in_sizes, int n_in,
                              void* d_out, int out_size, void* d_ws, size_t ws_size,
                              hipStream_t stream) {
    const float* inputs     = (const float*)d_in[0];
    const float* enc_pre_w0 = (const float*)d_in[8];
    const float* enc_pre_b0 = (const float*)d_in[9];
    const float* enc_pre_w1 = (const float*)d_in[10];
    const float* enc_pre_b1 = (const float*)d_in[11];
    const float* enc_wih    = (const float*)d_in[12];
    const float* enc_whh    = (const float*)d_in[13];
    const float* enc_bih    = (const float*)d_in[14];
    const float* enc_bhh    = (const float*)d_in[15];
    const float* dec_pre_w0 = (const float*)d_in[16];
    const float* dec_pre_b0 = (const float*)d_in[17];
    const float* dec_pre_w1 = (const float*)d_in[18];
    const float* dec_pre_b1 = (const float*)d_in[19];
    const float* dec_wih    = (const float*)d_in[20];
    const float* dec_whh    = (const float*)d_in[21];
    const float* dec_bih    = (const float*)d_in[22];
    const float* dec_bhh    = (const float*)d_in[23];
    const float* dec_post_w0 = (const float*)d_in[24];
    const float* dec_post_b0 = (const float*)d_in[25];
    const float* dec_post_w1 = (const float*)d_in[26];
    const float* dec_post_b1 = (const float*)d_in[27];
    const float* proj_w     = (const float*)d_in[28];
    const float* proj_b     = (const float*)d_in[29];
    float* out = (float*)d_out;

    float* ws = (float*)d_ws;
    const size_t LHN = (size_t)BN * HID;          // 2,662,400
    const size_t WT  = 2 * 262144;                // per transposed [2,256,1024] tensor
    size_t o = 0;
    float* W_enc0   = ws + o; o += 2 * HID;
    float* W_enc1   = ws + o; o += HID * HID;
    float* W_dpre0  = ws + o; o += HID;
    float* W_dpre1  = ws + o; o += HID * HID;
    float* W_dpost0 = ws + o; o += HID * HID;
    float* W_dpost1 = ws + o; o += HID * HID;
    float* enc_wihT = ws + o; o += WT;
    float* enc_whhT = ws + o; o += WT;
    float* dec_wihT = ws + o; o += WT;
    float* dec_whhT = ws + o; o += WT;
    float* enc_bias = ws + o; o += 2048;
    float* dec_bias = ws + o; o += 2048;
    float* xa       = ws + o; o += LHN;
    float* xb       = ws + o; o += LHN;
    float* h_a      = ws + o; o += 2 * LHN;
    float* h_b      = ws + o; o += 2 * LHN;
    float* c_st     = ws + o; o += 2 * LHN;
    float* dec_in   = ws + o; o += BN;

    // ---- init state (harness poisons ws; we must zero h0/c0/dec_in every call) ----
    {
        int n = (int)(2 * LHN);
        zero_kernel<<<(n + 255) / 256, 256, 0, stream>>>(h_a, n);
        zero_kernel<<<(n + 255) / 256, 256, 0, stream>>>(c_st, n);
        zero_kernel<<<(BN + 255) / 256, 256, 0, stream>>>(dec_in, BN);
    }

    // ---- weight prep ----
    sum_blocks_kernel<<<(2 * HID + 255) / 256, 256, 0, stream>>>(enc_pre_w0, W_enc0, 2);
    sum_blocks_kernel<<<256, 256, 0, stream>>>(enc_pre_w1, W_enc1, HID);
    sum_blocks_kernel<<<1, 256, 0, stream>>>(dec_pre_w0, W_dpre0, 1);
    sum_blocks_kernel<<<256, 256, 0, stream>>>(dec_pre_w1, W_dpre1, HID);
    sum_blocks_kernel<<<256, 256, 0, stream>>>(dec_post_w0, W_dpost0, HID);
    sum_blocks_kernel<<<256, 256, 0, stream>>>(dec_post_w1, W_dpost1, HID);
    transpose2_kernel<<<2048, 256, 0, stream>>>(enc_wih, enc_wihT);
    transpose2_kernel<<<2048, 256, 0, stream>>>(enc_whh, enc_whhT);
    transpose2_kernel<<<2048, 256, 0, stream>>>(dec_wih, dec_wihT);
    transpose2_kernel<<<2048, 256, 0, stream>>>(dec_whh, dec_whhT);
    add_bias_kernel<<<8, 256, 0, stream>>>(enc_bih, enc_bhh, enc_bias, 2048);
    add_bias_kernel<<<8, 256, 0, stream>>>(dec_bih, dec_bhh, dec_bias, 2048);

    dim3 gGemm(BN / 32, 4);   // 325 x 4 blocks, 256 threads (8 wave32)
    float* hcur = h_a;
    float* hnew = h_b;

    // ---- encoder: 12 time steps ----
    for (int t = 0; t < TSTEPS; ++t) {
        enc_l0_kernel<<<BN, 256, 0, stream>>>(inputs, t, W_enc0, enc_pre_b0, xa);
        gemm_relu_kernel<<<gGemm, 256, 0, stream>>>(xa, W_enc1, enc_pre_b1, xb);
        lstm_step_kernel<<<gGemm, 256, 0, stream>>>(xb, hcur, hnew, c_st,
                                                    enc_wihT, enc_whhT, enc_bias);
        lstm_step_kernel<<<gGemm, 256, 0, stream>>>(hnew, hcur + LHN, hnew + LHN, c_st + LHN,
                                                    enc_wihT + 262144, enc_whhT + 262144,
                                                    enc_bias + 1024);
        float* tmp = hcur; hcur = hnew; hnew = tmp;
    }

    // ---- autoregressive decoder: 12 steps (state carries over from encoder) ----
    for (int t = 0; t < TSTEPS; ++t) {
        dec_l0_kernel<<<BN, 256, 0, stream>>>(dec_in, W_dpre0, dec_pre_b0, xa);
        gemm_relu_kernel<<<gGemm, 256, 0, stream>>>(xa, W_dpre1, dec_pre_b1, xb);
        lstm_step_kernel<<<gGemm, 256, 0, stream>>>(xb, hcur, hnew, c_st,
                                                    dec_wihT, dec_whhT, dec_bias);
        lstm_step_kernel<<<gGemm, 256, 0, stream>>>(hnew, hcur + LHN, hnew + LHN, c_st + LHN,
                                                    dec_wihT + 262144, dec_whhT + 262144,
                                                    dec_bias + 1024);
        float* tmp = hcur; hcur = hnew; hnew = tmp;
        // xf = new h of layer 1 (now in hcur + LHN)
        gemm_relu_kernel<<<gGemm, 256, 0, stream>>>(hcur + LHN, W_dpost0, dec_post_b0, xa);
        gemm_relu_kernel<<<gGemm, 256, 0, stream>>>(xa, W_dpost1, dec_post_b1, xb);
        proj_kernel<<<BN / 8, 256, 0, stream>>>(xb, proj_w, proj_b, out, dec_in, t);
    }
}